// GraphSage_47433618817188
// MI455X (gfx1250) — compile-verified
//
#include <hip/hip_runtime.h>
#include <hip/hip_bf16.h>

typedef __attribute__((ext_vector_type(2))) float v2f;
typedef __attribute__((ext_vector_type(4))) float v4f;
typedef __attribute__((ext_vector_type(8))) float v8f;
typedef __attribute__((ext_vector_type(4))) int   v4i;

#define N_NODES 1000000
#define DOUT 128

#define GLOBAL_AS __attribute__((address_space(1)))
#define LDS_AS __attribute__((address_space(3)))

#if __has_builtin(__builtin_amdgcn_global_load_async_to_lds_b32) && \
    __has_builtin(__builtin_amdgcn_global_load_async_to_lds_b128)
#define HAVE_ASYNC_LDS 1
#else
#define HAVE_ASYNC_LDS 0
#endif

__device__ __forceinline__ void wait_asynccnt0() {
#if HAVE_ASYNC_LDS
#if __has_builtin(__builtin_amdgcn_s_wait_asynccnt)
  __builtin_amdgcn_s_wait_asynccnt(0);
#else
  asm volatile("s_wait_asynccnt 0" ::: "memory");
#endif
#endif
}

// ---------------------------------------------------------------------------
// Pack W[DOUT, K] (row-major) into K-pair-major Bp[KPAD/2][DOUT] of float2:
//   Bp[p*DOUT + n] = { W[n*K + 2p], W[n*K + 2p+1] }   (zero-padded past K)
// This makes each lane's WMMA B-fragment (two consecutive K rows, same N)
// a single contiguous global_load_b64 with immediate offsets.
// ---------------------------------------------------------------------------
__global__ void pack_w_pairs_kernel(const float* __restrict__ w,
                                    v2f* __restrict__ bp,
                                    int ktrue, int kpad) {
  int tid = blockIdx.x * blockDim.x + threadIdx.x;
  int total = (kpad / 2) * DOUT;
  if (tid >= total) return;
  int p = tid / DOUT;
  int n = tid % DOUT;
  v2f v;
  v.x = (2 * p     < ktrue) ? w[n * ktrue + 2 * p]     : 0.0f;
  v.y = (2 * p + 1 < ktrue) ? w[n * ktrue + 2 * p + 1] : 0.0f;
  bp[tid] = v;
}

// ---------------------------------------------------------------------------
// One GraphSAGE layer:
//   agg  = masked-mean over 3 sampled neighbors of feats[N, DIN]
//   out  = l2normalize(relu([self || agg] @ W^T))       out: [N, 128]
//
// Block = 128 threads = 4 wave32; each wave owns one 16-row node tile.
// GEMM: chained V_WMMA_F32_16X16X4_F32 (full fp32; the problem is memory-
// bound at ~24 FLOP/B so fp32 matrix throughput is free). A from LDS,
// B from the cache-resident pair-packed weights (one b64 per lane/step).
// Self-feature rows are copied global->LDS with GLOBAL_LOAD_ASYNC_TO_LDS.
// ---------------------------------------------------------------------------
template <int DIN, int KPAD>
__global__ __launch_bounds__(128) void sage_layer_kernel(
    const float* __restrict__ feats,   // [N, DIN]
    const v2f* __restrict__ bp,        // [KPAD/2, 128] pair-packed W^T
    const int* __restrict__ nidx,      // [N, 3]
    const int* __restrict__ nmask,     // [N, 3]
    float* __restrict__ out)           // [N, 128]
{
  constexpr int NSTEPS = KPAD / 4;

  __shared__ alignas(16) float comb[4][16][KPAD];   // [self || agg] tiles
  __shared__ alignas(16) float stage[4][16][DOUT];  // relu(GEMM) staging
  __shared__ float rnorm[4][16];                    // 1 / max(||row||, 1e-12)

  const int lane = threadIdx.x & 31;
  const int wv   = threadIdx.x >> 5;
  const int tile = blockIdx.x * 4 + wv;             // 16-row node tile

  // ---- gather phase: build combined[16][KPAD] = [self(DIN) | agg(DIN) | 0pad]
  for (int r = 0; r < 16; ++r) {
    const int row = tile * 16 + r;
    const int i0 = nidx[row * 3 + 0];
    const int i1 = nidx[row * 3 + 1];
    const int i2 = nidx[row * 3 + 2];
    const int m0 = nmask[row * 3 + 0];
    const int m1 = nmask[row * 3 + 1];
    const int m2 = nmask[row * 3 + 2];
    const int cnt = m0 + m1 + m2;
    const float inv_cnt = 1.0f / (float)(cnt > 0 ? cnt : 1);

    const float* xr = feats + (size_t)row * DIN;
    const float* x0 = feats + (size_t)i0 * DIN;
    const float* x1 = feats + (size_t)i1 * DIN;
    const float* x2 = feats + (size_t)i2 * DIN;

    // gfx1250 global_prefetch_b8 on the (random) neighbor rows
    __builtin_prefetch(x0 + lane, 0, 1);
    __builtin_prefetch(x1 + lane, 0, 1);
    __builtin_prefetch(x2 + lane, 0, 1);

    // self row: async global -> LDS (ASYNCcnt), no VGPR round-trip
    if constexpr (DIN % 128 == 0) {
#if HAVE_ASYNC_LDS
      for (int e = lane * 4; e < DIN; e += 128)
        __builtin_amdgcn_global_load_async_to_lds_b128(
            (GLOBAL_AS v4i*)(xr + e), (LDS_AS v4i*)&comb[wv][r][e], 0, 0);
#else
      for (int e = lane * 4; e < DIN; e += 128)
        *reinterpret_cast<v4f*>(&comb[wv][r][e]) =
            *reinterpret_cast<const v4f*>(xr + e);
#endif
    } else {
#if HAVE_ASYNC_LDS
      for (int e = lane; e < DIN; e += 32)
        __builtin_amdgcn_global_load_async_to_lds_b32(
            (GLOBAL_AS int*)(xr + e), (LDS_AS int*)&comb[wv][r][e], 0, 0);
#else
      for (int e = lane; e < DIN; e += 32) comb[wv][r][e] = xr[e];
#endif
    }

    // masked-mean aggregate (vectorized when DIN is a multiple of 4)
    if constexpr (DIN % 4 == 0) {
      for (int e = lane * 4; e < DIN; e += 128) {
        v4f s = {0.0f, 0.0f, 0.0f, 0.0f};
        if (m0) s += *reinterpret_cast<const v4f*>(x0 + e);
        if (m1) s += *reinterpret_cast<const v4f*>(x1 + e);
        if (m2) s += *reinterpret_cast<const v4f*>(x2 + e);
        *reinterpret_cast<v4f*>(&comb[wv][r][DIN + e]) = s * inv_cnt;
      }
    } else {
      for (int e = lane; e < DIN; e += 32) {
        float s = 0.0f;
        if (m0) s += x0[e];
        if (m1) s += x1[e];
        if (m2) s += x2[e];
        comb[wv][r][DIN + e] = s * inv_cnt;
      }
    }
    for (int e = 2 * DIN + lane; e < KPAD; e += 32) comb[wv][r][e] = 0.0f;
  }
  wait_asynccnt0();   // async LDS writes visible before WMMA reads the tile
  __syncthreads();

  // ---- WMMA phase: D[16,128] = comb[16,KPAD] @ W^T
  // A layout (16x4 f32): lanes 0-15 hold row M=lane, K={0,1}; lanes 16-31
  // hold M=lane-16, K={2,3}. B mirrors with lane%16 = N, half selects K pair.
  const int nl = lane & 15;
  const int hb = lane >> 4;       // K-pair half: 0 -> K{0,1}, 1 -> K{2,3}
  const int kh = hb << 1;

  for (int n0 = 0; n0 < 8; ++n0) {
    const v2f* bcol = bp + (size_t)hb * DOUT + n0 * 16 + nl;  // + s*2*DOUT
    v8f c = {0.0f, 0.0f, 0.0f, 0.0f, 0.0f, 0.0f, 0.0f, 0.0f};
#pragma unroll 4
    for (int s = 0; s < NSTEPS; ++s) {
      // A fragment: two consecutive K values of this lane's row (ds_load_b64)
      v2f av = *reinterpret_cast<const v2f*>(&comb[wv][nl][4 * s + kh]);
      // B fragment: one contiguous pair-packed global_load_b64
      v2f bv = bcol[(size_t)2 * s * DOUT];
      c = __builtin_amdgcn_wmma_f32_16x16x4_f32(
          /*neg_a=*/false, av, /*neg_b=*/false, bv,
          /*c_mod=*/(short)0, c, /*reuse_a=*/false, /*reuse_b=*/false);
    }
    // ReLU + stage; C layout: VGPR i -> row i + 8*(lane/16), col = lane%16
#pragma unroll
    for (int i = 0; i < 8; ++i) {
      float v = c[i] > 0.0f ? c[i] : 0.0f;
      stage[wv][i + (hb << 3)][n0 * 16 + nl] = v;
    }
  }
  __syncthreads();

  // ---- row L2 norms (lanes 0-15, one row each)
  if (lane < 16) {
    float ss = 0.0f;
#pragma unroll 8
    for (int e = 0; e < DOUT; ++e) {
      float v = stage[wv][lane][e];
      ss += v * v;
    }
    float nrm = sqrtf(ss);
    rnorm[wv][lane] = 1.0f / fmaxf(nrm, 1e-12f);
  }
  __syncthreads();

  // ---- normalized, coalesced stores
  for (int m = 0; m < 16; ++m) {
    const size_t row = (size_t)(tile * 16 + m);
    const float sc = rnorm[wv][m];
    for (int e = lane; e < DOUT; e += 32)
      out[row * DOUT + e] = stage[wv][m][e] * sc;
  }
}

// ---------------------------------------------------------------------------
extern "C" void kernel_launch(void* const* d_in, const int* in_sizes, int n_in,
                              void* d_out, int out_size, void* d_ws, size_t ws_size,
                              hipStream_t stream) {
  const float* x   = (const float*)d_in[0];  // [N, 57]
  const float* W1  = (const float*)d_in[1];  // [128, 114]
  const float* W2  = (const float*)d_in[2];  // [128, 256]
  const int*   ni1 = (const int*)d_in[3];    // [N, 3]
  const int*   nm1 = (const int*)d_in[4];    // [N, 3]
  const int*   ni2 = (const int*)d_in[5];    // [N, 3]
  const int*   nm2 = (const int*)d_in[6];    // [N, 3]
  float* out = (float*)d_out;                // [N, 128]

  // workspace layout: h1 [N,128] | Bp1 [58,128] float2 | Bp2 [128,128] float2
  float* h1  = (float*)d_ws;
  v2f* bp1 = (v2f*)(h1 + (size_t)N_NODES * DOUT);
  v2f* bp2 = bp1 + (size_t)(116 / 2) * DOUT;

  pack_w_pairs_kernel<<<((116 / 2) * DOUT + 255) / 256, 256, 0, stream>>>(W1, bp1, 114, 116);
  pack_w_pairs_kernel<<<((256 / 2) * DOUT + 255) / 256, 256, 0, stream>>>(W2, bp2, 256, 256);

  dim3 grid(N_NODES / 64);  // 15625 blocks x 4 waves x 16 rows = 1M rows
  sage_layer_kernel<57, 116><<<grid, 128, 0, stream>>>(x, bp1, ni1, nm1, h1);
  sage_layer_kernel<128, 256><<<grid, 128, 0, stream>>>(h1, bp2, ni2, nm2, out);
}